// GcnNet_7576322310702
// MI455X (gfx1250) — compile-verified
//
#include <hip/hip_runtime.h>
#include <hip/hip_bf16.h>

typedef float v2f __attribute__((ext_vector_type(2)));
typedef float v8f __attribute__((ext_vector_type(8)));

// ---------------------------------------------------------------- init
// deg[i] = 1 (self loop), agg2[:] = 0
__global__ void k_init(float* __restrict__ deg, float* __restrict__ agg2, int n) {
    int i = blockIdx.x * blockDim.x + threadIdx.x;
    if (i < n * 64) agg2[i] = 0.0f;
    if (i < n)      deg[i]  = 1.0f;
}

// ---------------------------------------------------------------- degree
__global__ void k_deg(const int* __restrict__ dst, float* __restrict__ deg, int E) {
    int e = blockIdx.x * blockDim.x + threadIdx.x;
    if (e < E) atomicAdd(&deg[dst[e]], 1.0f);
}

// ---------------------------------------------------------------- dis + layer-1 self term
// dis[i] = rsqrt(deg[i]) (deg >= 1 always via self loops); a1[i] = x[i]*dis[i]^2
__global__ void k_dis(float* __restrict__ deg_dis, float* __restrict__ a1,
                      const float* __restrict__ x, int n) {
    int i = blockIdx.x * blockDim.x + threadIdx.x;
    if (i >= n) return;
    float dis = __frsqrt_rn(deg_dis[i]);
    deg_dis[i] = dis;
    a1[i] = x[i] * dis * dis;
}

// ---------------------------------------------------------------- layer-1 scalar edge scatter
// rank-1 trick: aggregate scalar x[s]*norm; projection by W1 happens inside the GEMM
__global__ void k_edge1(const int* __restrict__ src, const int* __restrict__ dst,
                        const float* __restrict__ x, const float* __restrict__ dis,
                        float* __restrict__ a1, int E) {
    int e = blockIdx.x * blockDim.x + threadIdx.x;
    if (e >= E) return;
    int s = src[e], d = dst[e];
    atomicAdd(&a1[d], x[s] * dis[s] * dis[d]);
}

// ---------------------------------------------------------------- fused layer-1 epilogue + WMMA GEMM
// h[N,64] = relu(a1 ⊗ W1 + b1) @ W2.  One wave32 owns a 16-row x 64-col slab:
// 4 col-tiles x 8 K-steps of v_wmma_f32_16x16x4_f32 (32 WMMAs/wave).
// A-fragment is built in-register: A[row][k] = relu(a1[row]*W1[k] + b1[k]).
__global__ void k_gemm(const float* __restrict__ a1, const float* __restrict__ W1,
                       const float* __restrict__ b1, const float* __restrict__ W2,
                       float* __restrict__ h, int n) {
    int tid  = blockIdx.x * blockDim.x + threadIdx.x;
    int wave = tid >> 5;
    int lane = threadIdx.x & 31;
    int base = wave * 16;
    if (base >= n) return;              // wave-uniform exit, EXEC stays full

    int half = lane >> 4;               // 0: K k..k+1 / rows M..M+7 ; 1: K k+2..k+3 / rows M+8..
    int col  = lane & 15;
    int arow = base + col;
    if (arow >= n) arow = n - 1;        // data-level clamp, EXEC unaffected

    float av = a1[arow];

    // A fragments (16x4 f32 per K-step), ISA 7.12.2 layout
    v2f a[8];
#pragma unroll
    for (int kt = 0; kt < 8; ++kt) {
        int k = kt * 4 + half * 2;
        float v0 = av * W1[k]     + b1[k];
        float v1 = av * W1[k + 1] + b1[k + 1];
        a[kt].x = fmaxf(v0, 0.0f);
        a[kt].y = fmaxf(v1, 0.0f);
    }

    bool full = (base + 16 <= n);       // one branch instead of 32 per-store guards
#pragma unroll
    for (int ct = 0; ct < 4; ++ct) {
        v8f acc = {};
#pragma unroll
        for (int kt = 0; kt < 8; ++kt) {
            int kb = kt * 4 + half * 2;
            v2f b;
            b.x = W2[(kb    ) * 64 + ct * 16 + col];
            b.y = W2[(kb + 1) * 64 + ct * 16 + col];
            acc = __builtin_amdgcn_wmma_f32_16x16x4_f32(
                false, a[kt], false, b, (short)0, acc, false, false);
        }
        // D layout: lane l, vgpr v -> row = base + v + 8*(l/16), col = ct*16 + l%16
        float* hp = h + (base + half * 8) * 64 + ct * 16 + col;
        if (full) {
#pragma unroll
            for (int v = 0; v < 8; ++v) hp[v * 64] = acc[v];
        } else {
#pragma unroll
            for (int v = 0; v < 8; ++v) {
                if (base + half * 8 + v < n) hp[v * 64] = acc[v];
            }
        }
    }
}

// ---------------------------------------------------------------- layer-2 edge scatter
// 16 lanes per edge, 4 features each. Leader lane (0/16) loads indices + norm,
// broadcast via wave32 shuffle; float4 gather + 4 f32 atomics per lane.
__global__ void k_edge2(const int* __restrict__ src, const int* __restrict__ dst,
                        const float* __restrict__ dis, const float* __restrict__ h,
                        float* __restrict__ agg2, int E) {
    int tid  = blockIdx.x * blockDim.x + threadIdx.x;
    int lane = threadIdx.x & 31;
    int sub  = lane & 15;
    int e    = tid >> 4;
    if (e >= E) return;

    int s = 0, d = 0;
    float norm = 0.0f;
    if (sub == 0) {                     // one load set per 16-lane group
        s = src[e];
        d = dst[e];
        norm = dis[s] * dis[d];
    }
    int leader = lane & 16;             // lane 0 or lane 16
    s    = __shfl(s, leader);
    d    = __shfl(d, leader);
    norm = __shfl(norm, leader);

    const float4 hv = *(const float4*)(h + s * 64 + sub * 4);
    float* op = agg2 + d * 64 + sub * 4;
    atomicAdd(op + 0, hv.x * norm);
    atomicAdd(op + 1, hv.y * norm);
    atomicAdd(op + 2, hv.z * norm);
    atomicAdd(op + 3, hv.w * norm);
}

// ---------------------------------------------------------------- epilogue
// out[i] = relu(agg2[i] + h[i]*dis[i]^2 + b2) . Wf + bf   (float4 / b128 loads)
__global__ void k_final(const float* __restrict__ agg2, const float* __restrict__ h,
                        const float* __restrict__ dis, const float* __restrict__ b2,
                        const float* __restrict__ Wf, const float* __restrict__ bf,
                        float* __restrict__ out, int n) {
    int i = blockIdx.x * blockDim.x + threadIdx.x;
    if (i >= n) return;
    float sd  = dis[i] * dis[i];
    float acc = bf[0];
    const float4* ag4 = (const float4*)(agg2 + i * 64);
    const float4* hp4 = (const float4*)(h + i * 64);
    const float4* b24 = (const float4*)b2;
    const float4* wf4 = (const float4*)Wf;
#pragma unroll
    for (int j = 0; j < 16; ++j) {
        float4 ag = ag4[j];
        float4 hp = hp4[j];
        float4 bb = b24[j];
        float4 wf = wf4[j];
        acc += fmaxf(ag.x + hp.x * sd + bb.x, 0.0f) * wf.x;
        acc += fmaxf(ag.y + hp.y * sd + bb.y, 0.0f) * wf.y;
        acc += fmaxf(ag.z + hp.z * sd + bb.z, 0.0f) * wf.z;
        acc += fmaxf(ag.w + hp.w * sd + bb.w, 0.0f) * wf.w;
    }
    out[i] = acc;
}

// ---------------------------------------------------------------- launcher
extern "C" void kernel_launch(void* const* d_in, const int* in_sizes, int n_in,
                              void* d_out, int out_size, void* d_ws, size_t ws_size,
                              hipStream_t stream) {
    const float* x   = (const float*)d_in[0];
    const int*   ei  = (const int*)d_in[1];     // [2, E] flat: src then dst
    const float* W1  = (const float*)d_in[2];
    const float* b1  = (const float*)d_in[3];
    const float* W2  = (const float*)d_in[4];
    const float* b2  = (const float*)d_in[5];
    const float* Wf  = (const float*)d_in[6];
    const float* bf  = (const float*)d_in[7];
    float*       out = (float*)d_out;

    const int n = in_sizes[0];          // 100000
    const int E = in_sizes[1] / 2;      // 1000000
    const int* src = ei;
    const int* dst = ei + E;

    // workspace layout (floats)
    float* ws   = (float*)d_ws;
    float* dis  = ws;                   // [n]    deg -> dis (in place)
    float* a1   = ws + n;               // [n]    layer-1 scalar aggregate
    float* h    = ws + 2 * n;           // [n*64] post-GEMM features
    float* agg2 = ws + 66 * n;          // [n*64] layer-2 aggregate

    const int B = 256;
    auto cdiv = [](int a, int b) { return (a + b - 1) / b; };

    k_init <<<cdiv(n * 64, B), B, 0, stream>>>(dis, agg2, n);
    k_deg  <<<cdiv(E, B), B, 0, stream>>>(dst, dis, E);
    k_dis  <<<cdiv(n, B), B, 0, stream>>>(dis, a1, x, n);
    k_edge1<<<cdiv(E, B), B, 0, stream>>>(src, dst, x, dis, a1, E);

    int waves = cdiv(n, 16);
    k_gemm <<<cdiv(waves * 32, B), B, 0, stream>>>(a1, W1, b1, W2, h, n);

    k_edge2<<<cdiv(E * 16, B), B, 0, stream>>>(src, dst, dis, h, agg2, E);
    k_final<<<cdiv(n, B), B, 0, stream>>>(agg2, h, dis, b2, Wf, bf, out, n);
}